// LinearAttention_35184372089345
// MI455X (gfx1250) — compile-verified
//
#include <hip/hip_runtime.h>
#include <hip/hip_bf16.h>

// Linear attention, fp32, CDNA5 (gfx1250, wave32).
//   Pass 1: KV[b] = phi(K)^T V  (64x64), Ksum[b] = sum_n phi(K[n])  (64)
//   Pass 2: out[n] = (phi(Q[n]) @ KV) / (phi(Q[n]) . Ksum + eps)
// Matmuls use V_WMMA_F32_16X16X4_F32 (full fp32 precision, f32 accumulate).
// Pass 1 chunk loop unrolling is capped so VGPR usage stays under 256
// (avoids s_set_vgpr_msb churn and keeps occupancy for latency hiding).

typedef __attribute__((ext_vector_type(2))) float v2f;
typedef __attribute__((ext_vector_type(8))) float v8f;

#define BN   32
#define NTOK 8192
#define CDIM 64
#define DDIM 64
#define EPSV 1e-6f

#define P1_SPLITS 32              // N-splits per batch in pass 1
#define P1_TOK_PER_BLOCK (NTOK / P1_SPLITS)     // 256
#define P1_TOK_PER_WAVE  (P1_TOK_PER_BLOCK / 8) // 32 -> 8 chunks of 4

__device__ __forceinline__ float phi_elu1(float x) {
    // elu(x) + 1 == (x > 0) ? x + 1 : exp(x)
    return x > 0.0f ? x + 1.0f : __expf(x);
}

// ---------------------------------------------------------------- zero init
__global__ void la_zero_kernel(float* __restrict__ p, int n) {
    int i = blockIdx.x * blockDim.x + threadIdx.x;
    if (i < n) p[i] = 0.0f;
}

// ---------------------------------------------------------------- pass 1
// grid = BN * P1_SPLITS blocks, 256 threads (8 waves).
// Each wave: 32 tokens in chunks of 4; per chunk 16 WMMA (4 c-tiles x 4 d-tiles).
__global__ __launch_bounds__(256) void la_pass1_kernel(
    const float* __restrict__ K, const float* __restrict__ V,
    float* __restrict__ kv_g, float* __restrict__ ksum_g) {

    __shared__ float kv_lds[CDIM * DDIM];
    __shared__ float ksum_lds[CDIM];

    const int tid = threadIdx.x;
    for (int i = tid; i < CDIM * DDIM; i += 256) kv_lds[i] = 0.0f;
    if (tid < CDIM) ksum_lds[tid] = 0.0f;
    __syncthreads();

    const int b     = blockIdx.x / P1_SPLITS;
    const int split = blockIdx.x % P1_SPLITS;
    const int wave  = tid >> 5;
    const int lane  = tid & 31;
    const int h     = lane >> 4;   // half: 0 -> K=0,1 ; 1 -> K=2,3 (ISA A layout)
    const int l     = lane & 15;

    const float* __restrict__ Kb = K + (size_t)b * NTOK * CDIM;
    const float* __restrict__ Vb = V + (size_t)b * NTOK * DDIM;

    v8f acc[4][4];
    const v8f vzero = {};
#pragma unroll
    for (int ct = 0; ct < 4; ++ct)
#pragma unroll
        for (int dt = 0; dt < 4; ++dt) acc[ct][dt] = vzero;
    float ks_acc[4] = {0.0f, 0.0f, 0.0f, 0.0f};

    const int n_base = split * P1_TOK_PER_BLOCK + wave * P1_TOK_PER_WAVE;

    // Cap unrolling: 128 acc VGPRs are live across the whole loop; full
    // unroll (8x) pushed total usage past 256 and triggered VGPR-MSB mode.
#pragma unroll 2
    for (int chunk = 0; chunk < P1_TOK_PER_WAVE / 4; ++chunk) {
        // this lane's token pair inside the 4-token K-slab
        const int n0 = n_base + chunk * 4 + 2 * h;

        v2f afr[4], bfr[4];
        // A fragments: A[M=c][K=token], M tile = ct*16 + l
#pragma unroll
        for (int ct = 0; ct < 4; ++ct) {
            const int c = ct * 16 + l;
            const float k0 = phi_elu1(Kb[(size_t)n0 * CDIM + c]);
            const float k1 = phi_elu1(Kb[(size_t)(n0 + 1) * CDIM + c]);
            afr[ct].x = k0;
            afr[ct].y = k1;
            ks_acc[ct] += k0 + k1;      // column sums of phi(K)
        }
        // B fragments: B[K=token][N=d], N tile = dt*16 + l
#pragma unroll
        for (int dt = 0; dt < 4; ++dt) {
            const int d = dt * 16 + l;
            bfr[dt].x = Vb[(size_t)n0 * DDIM + d];
            bfr[dt].y = Vb[(size_t)(n0 + 1) * DDIM + d];
        }
#pragma unroll
        for (int ct = 0; ct < 4; ++ct)
#pragma unroll
            for (int dt = 0; dt < 4; ++dt)
                acc[ct][dt] = __builtin_amdgcn_wmma_f32_16x16x4_f32(
                    false, afr[ct], false, bfr[dt], (short)0, acc[ct][dt],
                    false, false);
    }

    // wave -> LDS reduction (C/D layout: vgpr i holds row i + 8*h, col l)
#pragma unroll
    for (int ct = 0; ct < 4; ++ct) {
#pragma unroll
        for (int dt = 0; dt < 4; ++dt) {
#pragma unroll
            for (int i = 0; i < 8; ++i) {
                const int row = ct * 16 + i + 8 * h;
                const int col = dt * 16 + l;
                atomicAdd(&kv_lds[row * DDIM + col], acc[ct][dt][i]);
            }
        }
        const float tot = ks_acc[ct] + __shfl_xor(ks_acc[ct], 16, 32);
        if (h == 0) atomicAdd(&ksum_lds[ct * 16 + l], tot);
    }
    __syncthreads();

    // block -> global partial reduction
    float* __restrict__ kvb = kv_g + (size_t)b * CDIM * DDIM;
    for (int i = tid; i < CDIM * DDIM; i += 256) atomicAdd(&kvb[i], kv_lds[i]);
    if (tid < CDIM) atomicAdd(&ksum_g[b * CDIM + tid], ksum_lds[tid]);
}

// ---------------------------------------------------------------- pass 2
// grid = BN*NTOK/128 blocks, 256 threads (8 waves); wave = 16 tokens x 64 outs.
__global__ __launch_bounds__(256) void la_pass2_kernel(
    const float* __restrict__ Q, const float* __restrict__ kv_g,
    const float* __restrict__ ksum_g, float* __restrict__ out) {

    __shared__ float kv_lds[CDIM * DDIM];
    __shared__ float ksum_lds[CDIM];

    const int tid = threadIdx.x;
    const int b = (int)(((size_t)blockIdx.x * 128) / NTOK);

    for (int i = tid; i < CDIM * DDIM; i += 256)
        kv_lds[i] = kv_g[(size_t)b * CDIM * DDIM + i];
    if (tid < CDIM) ksum_lds[tid] = ksum_g[b * CDIM + tid];
    __syncthreads();

    const int wave = tid >> 5;
    const int lane = tid & 31;
    const int h = lane >> 4;
    const int l = lane & 15;
    const int n_base = (blockIdx.x * 128 + wave * 16) % NTOK;

    const float* __restrict__ Qb = Q + (size_t)b * NTOK * CDIM;
    float* __restrict__ Ob = out + (size_t)b * NTOK * DDIM;

    v8f acc[4];
    const v8f vzero = {};
#pragma unroll
    for (int dt = 0; dt < 4; ++dt) acc[dt] = vzero;

    float den = 0.0f;
    const float* __restrict__ qrow = Qb + (size_t)(n_base + l) * CDIM;

#pragma unroll 4
    for (int c0 = 0; c0 < CDIM; c0 += 4) {
        const int ks = c0 + 2 * h;            // this half's two K indices
        v2f a;
        a.x = phi_elu1(qrow[ks]);
        a.y = phi_elu1(qrow[ks + 1]);
        den += a.x * ksum_lds[ks] + a.y * ksum_lds[ks + 1];

#pragma unroll
        for (int dt = 0; dt < 4; ++dt) {
            v2f bf;
            bf.x = kv_lds[ks * DDIM + dt * 16 + l];
            bf.y = kv_lds[(ks + 1) * DDIM + dt * 16 + l];
            acc[dt] = __builtin_amdgcn_wmma_f32_16x16x4_f32(
                false, a, false, bf, (short)0, acc[dt], false, false);
        }
    }

    // full denominator for token n_base + (lane & 15)
    den += __shfl_xor(den, 16, 32);

#pragma unroll
    for (int i = 0; i < 8; ++i) {
        const int row = i + 8 * h;                 // C/D layout row for vgpr i
        const float dr = __shfl(den, row, 32);     // denom of that token
        const float z = 1.0f / (dr + EPSV);
#pragma unroll
        for (int dt = 0; dt < 4; ++dt)
            Ob[(size_t)(n_base + row) * DDIM + dt * 16 + l] = acc[dt][i] * z;
    }
}

// ---------------------------------------------------------------- launch
extern "C" void kernel_launch(void* const* d_in, const int* in_sizes, int n_in,
                              void* d_out, int out_size, void* d_ws, size_t ws_size,
                              hipStream_t stream) {
    const float* Q = (const float*)d_in[0];
    const float* K = (const float*)d_in[1];
    const float* V = (const float*)d_in[2];
    float* out = (float*)d_out;

    // workspace: KV [B][64][64] then Ksum [B][64]  (~0.52 MB)
    float* kv_g   = (float*)d_ws;
    float* ksum_g = kv_g + (size_t)BN * CDIM * DDIM;
    const int nws = BN * CDIM * DDIM + BN * CDIM;

    la_zero_kernel<<<(nws + 255) / 256, 256, 0, stream>>>(kv_g, nws);
    la_pass1_kernel<<<BN * P1_SPLITS, 256, 0, stream>>>(K, V, kv_g, ksum_g);
    la_pass2_kernel<<<(BN * NTOK) / 128, 256, 0, stream>>>(Q, kv_g, ksum_g, out);
}